// OneHotConv3d_8272107012225
// MI455X (gfx1250) — compile-verified
//
#include <hip/hip_runtime.h>

// OneHotConv3d: out[b,c,t,h,w] = bias[c] + sum_{dt,dh,dw} Wt[k][idx_clamped][c]
//   B=4, T=16, H=128, W=128, C_out=128, num_classes=256, k=(dt*3+dh)*3+dw
// Gather-bound op: 13.8 GB of L2-resident table gathers + 537 MB NT stores.

#define NCLS   256
#define COUT   128
#define T_DIM  16
#define H_DIM  128
#define W_DIM  128
#define NTAP   27
#define HALO_ROWS 9              // 3 dt x 3 dh
#define HALO_W    (W_DIM + 2)    // w' = -1 .. 128
#define HALO_N    (HALO_ROWS * HALO_W)   // 1170 staged indices
#define BLOCK  512

// ---- CDNA5 async global->LDS path (guarded; fallback is plain loads) ----
#if defined(__HIP_DEVICE_COMPILE__) && defined(__gfx1250__) && \
    __has_builtin(__builtin_amdgcn_global_load_async_to_lds_b32) && \
    __has_builtin(__builtin_amdgcn_s_wait_asynccnt)
#define USE_ASYNC_LDS 1
#else
#define USE_ASYNC_LDS 0
#endif

#if USE_ASYNC_LDS
typedef __attribute__((address_space(1))) int as1_int;
typedef __attribute__((address_space(3))) int as3_int;
#endif

// Repack weight (C,V,3,3,3) row-major -> Wt[k][v][c] (contiguous c, 512B rows)
__global__ void onehot_repack_kernel(const float* __restrict__ w,
                                     float* __restrict__ wt) {
  int i = blockIdx.x * blockDim.x + threadIdx.x;
  if (i >= NTAP * NCLS * COUT) return;
  int c = i & (COUT - 1);
  int v = (i >> 7) & (NCLS - 1);
  int k = i >> 15;                       // < 27
  wt[i] = w[(c * NCLS + v) * NTAP + k];
}

__global__ __launch_bounds__(BLOCK)
void onehot_conv3d_kernel(const long long* __restrict__ idx,
                          const float* __restrict__ wt,
                          const float* __restrict__ bias,
                          float* __restrict__ out) {
  __shared__ unsigned sIdx[HALO_N];

  const int h = blockIdx.x;
  const int t = blockIdx.y;
  const int b = blockIdx.z;
  const int tx = threadIdx.x;

  // ---- Stage edge-clamped 3x3x(W+2) index halo into LDS (async on CDNA5) ----
  const long long* idxB = idx + (long long)b * (T_DIM * H_DIM * W_DIM);
  for (int e = tx; e < HALO_N; e += BLOCK) {
    int r  = e / HALO_W;                 // 0..8  -> (dt,dh)
    int wi = e - r * HALO_W;             // 0..129
    int dt = r / 3;
    int dh = r - dt * 3;
    int ts = t + dt - 2; ts = ts < 0 ? 0 : ts;                       // causal T pad
    int hs = h + dh - 1; hs = hs < 0 ? 0 : (hs > H_DIM - 1 ? H_DIM - 1 : hs);
    int ws = wi - 1;     ws = ws < 0 ? 0 : (ws > W_DIM - 1 ? W_DIM - 1 : ws);
    const long long* gp = idxB + ((ts * H_DIM) + hs) * W_DIM + ws;
#if USE_ASYNC_LDS
    // int64 little-endian: b32 grabs the low dword (value < 256)
    __builtin_amdgcn_global_load_async_to_lds_b32(
        (as1_int*)gp, (as3_int*)&sIdx[e], 0, 0);
#else
    sIdx[e] = (unsigned)(*gp);
#endif
  }
#if USE_ASYNC_LDS
  __builtin_amdgcn_s_wait_asynccnt(0);
#endif
  __syncthreads();

  // ---- Layout: wave lanes span w (coalesced stores); 4 w-groups x 4 c-groups ----
  const int wave = tx >> 5;
  const int lane = tx & 31;
  const int w    = (wave & 3) * 32 + lane;   // 0..127
  const int cb   = (wave >> 2) * 32;         // 0,32,64,96

  float4 acc[8];
#pragma unroll
  for (int j = 0; j < 8; ++j) acc[j] = make_float4(0.f, 0.f, 0.f, 0.f);

#pragma unroll
  for (int dt = 0; dt < 3; ++dt) {
#pragma unroll
    for (int dh = 0; dh < 3; ++dh) {
      const int r = dt * 3 + dh;
      const unsigned* srow = &sIdx[r * HALO_W + w];
#pragma unroll
      for (int dw = 0; dw < 3; ++dw) {
        const unsigned v = srow[dw];                 // ds_load_b32, stride-1 -> no conflicts
        const int k = r * 3 + dw;
        // 128B line-aligned chunk of the (k,v) row
        const float4* row =
            (const float4*)(wt + ((k << 8) + (int)v) * COUT + cb);
#pragma unroll
        for (int j = 0; j < 8; ++j) {
          float4 wv = row[j];
          acc[j].x += wv.x; acc[j].y += wv.y;
          acc[j].z += wv.z; acc[j].w += wv.w;
        }
      }
    }
  }

  // ---- bias + non-temporal stores (output is write-once; keep table hot in L2) ----
  const int cs = T_DIM * H_DIM * W_DIM;    // channel stride = 262144
  float* outp = out +
      ((((long long)b * COUT + cb) * T_DIM + t) * H_DIM + h) * (long long)W_DIM + w;
  const float4* bp = (const float4*)(bias + cb);
#pragma unroll
  for (int j = 0; j < 8; ++j) {
    float4 bv = bp[j];
    __builtin_nontemporal_store(acc[j].x + bv.x, outp + (j * 4 + 0) * cs);
    __builtin_nontemporal_store(acc[j].y + bv.y, outp + (j * 4 + 1) * cs);
    __builtin_nontemporal_store(acc[j].z + bv.z, outp + (j * 4 + 2) * cs);
    __builtin_nontemporal_store(acc[j].w + bv.w, outp + (j * 4 + 3) * cs);
  }
}

extern "C" void kernel_launch(void* const* d_in, const int* in_sizes, int n_in,
                              void* d_out, int out_size, void* d_ws, size_t ws_size,
                              hipStream_t stream) {
  const long long* idx   = (const long long*)d_in[0];   // (B,T,H,W) int64
  const float*     weight= (const float*)d_in[1];       // (128,256,3,3,3) f32
  const float*     bias  = (const float*)d_in[2];       // (128,) f32
  float*           out   = (float*)d_out;               // (B,128,T,H,W) f32

  // scratch: repacked table Wt[27][256][128] f32 = 3.46 MB
  float* wt = (float*)d_ws;

  const int B = in_sizes[0] / (T_DIM * H_DIM * W_DIM);  // = 4

  {
    int n = NTAP * NCLS * COUT;
    onehot_repack_kernel<<<(n + 255) / 256, 256, 0, stream>>>(weight, wt);
  }
  {
    dim3 grid(H_DIM, T_DIM, B);
    onehot_conv3d_kernel<<<grid, BLOCK, 0, stream>>>(idx, wt, bias, out);
  }
}